// TensorProduct_88132728914170
// MI455X (gfx1250) — compile-verified
//
#include <hip/hip_runtime.h>
#include <stdint.h>

#define MUL   64
#define DIM1  576
#define DIM2  9
#define WDIM  960
#define ODIM  3264
#define ZPB   4
#define NTHR  (ZPB * MUL)   // 256 threads = 8 wave32

typedef __attribute__((ext_vector_type(4))) unsigned int u32x4;
typedef __attribute__((ext_vector_type(8))) int          i32x8;
typedef __attribute__((ext_vector_type(4))) int          i32x4;

// ---------------------------------------------------------------------------
// TDM: 1-D tile of nElems f32 from global -> LDS. Descriptor per CDNA5 ISA
// CDNA5 ISA: group0 = {count/type/lds/global addr},
// group1 = {data_size, tensor dims, tile dims, strides}. groups 2/3 unused.
// clang-23 lane: 6-arg builtin (extra int32x8 before cpol).
// ---------------------------------------------------------------------------
#if __has_builtin(__builtin_amdgcn_tensor_load_to_lds)
#define HAVE_TDM 1
__device__ __forceinline__ void tdm_load_1d(const void* gptr, unsigned lds_byte_off,
                                            unsigned nElems) {
  uint64_t ga = (uint64_t)(uintptr_t)gptr;
  u32x4 g0;
  g0.x = 1u;                                   // count=1, is_restore=0, gather off
  g0.y = lds_byte_off;                         // lds_addr (bytes)
  g0.z = (unsigned)(ga & 0xFFFFFFFFu);         // global_addr[31:0]
  g0.w = (unsigned)((ga >> 32) & 0x01FFFFFFu)  // global_addr[56:32]
       | 0x80000000u;                          // type = 2 ("image")
  i32x8 g1;
  g1[0] = (2 << 16);                           // wg_mask=0 | data_size=2 (4B)
  g1[1] = (int)((nElems & 0xFFFFu) << 16);     // tensor_dim0[15:0] @ bits 63:48
  g1[2] = (int)((nElems >> 16) & 0xFFFFu)      // tensor_dim0[31:16]
        | (1 << 16);                           // tensor_dim1 = 1
  g1[3] = (int)((nElems & 0xFFFFu) << 16);     // tile_dim0 = nElems @ bits 127:112
  g1[4] = 1;                                   // tile_dim1 = 1, tile_dim2 = 0
  g1[5] = (int)nElems;                         // tensor_dim0_stride[31:0]
  g1[6] = 0;                                   // stride0 hi, stride1 lo
  g1[7] = 0;
  i32x4 z4 = {0, 0, 0, 0};
  i32x8 z8 = {0, 0, 0, 0, 0, 0, 0, 0};
  __builtin_amdgcn_tensor_load_to_lds(g0, g1, z4, z4, z8, 0);
}
#else
#define HAVE_TDM 0
#endif

__launch_bounds__(NTHR)
__global__ void tp_kernel(const float* __restrict__ x1, const float* __restrict__ x2,
                          const float* __restrict__ wgt, float* __restrict__ out, int n) {
  __shared__ float sx1[ZPB * DIM1];
  __shared__ float sw [ZPB * WDIM];
  __shared__ float sx2[ZPB * DIM2];

  const int z0 = blockIdx.x * ZPB;
  const int zv = (n - z0) < ZPB ? (n - z0) : ZPB;   // valid z in this block (==ZPB normally)

#if HAVE_TDM
  if (threadIdx.x < 32) {  // one wave drives the Tensor Data Mover
    tdm_load_1d(x1 + (size_t)z0 * DIM1, (unsigned)(uintptr_t)&sx1[0], (unsigned)(zv * DIM1));
    tdm_load_1d(wgt + (size_t)z0 * WDIM, (unsigned)(uintptr_t)&sw[0],  (unsigned)(zv * WDIM));
    tdm_load_1d(x2 + (size_t)z0 * DIM2, (unsigned)(uintptr_t)&sx2[0], (unsigned)(zv * DIM2));
    __builtin_amdgcn_s_wait_tensorcnt(0);
  }
#else
  for (int i = threadIdx.x; i < zv * DIM1; i += NTHR) sx1[i] = x1[(size_t)z0 * DIM1 + i];
  for (int i = threadIdx.x; i < zv * WDIM; i += NTHR) sw[i]  = wgt[(size_t)z0 * WDIM + i];
  for (int i = threadIdx.x; i < zv * DIM2; i += NTHR) sx2[i] = x2[(size_t)z0 * DIM2 + i];
#endif
  __syncthreads();

  const int u  = threadIdx.x & (MUL - 1);
  const int zl = threadIdx.x >> 6;
  const int z  = z0 + zl;
  if (zl >= zv) return;

  const float* X1 = sx1 + zl * DIM1;
  const float* W  = sw  + zl * WDIM;
  const float* V  = sx2 + zl * DIM2;

  // x1: l=0 scalar, l=1 vector (y,z,x), l=2 (xy,yz,2z2-r2,xz,x2-y2)
  const float s  = X1[u];
  const float p0 = X1[64 + u * 3 + 0], p1 = X1[64 + u * 3 + 1], p2 = X1[64 + u * 3 + 2];
  const float d0 = X1[256 + u * 5 + 0], d1 = X1[256 + u * 5 + 1], d2 = X1[256 + u * 5 + 2],
              d3 = X1[256 + u * 5 + 3], d4 = X1[256 + u * 5 + 4];
  const float vs = V[0];
  const float q0 = V[1], q1 = V[2], q2 = V[3];
  const float e0 = V[4], e1 = V[5], e2 = V[6], e3 = V[7], e4 = V[8];

  const float w0 = W[u],          w1 = W[64 + u],   w2 = W[128 + u],  w3 = W[192 + u];
  const float w4 = W[256 + u],    w5 = W[320 + u],  w6 = W[384 + u],  w7 = W[448 + u];
  const float w8 = W[512 + u],    w9 = W[576 + u],  w10 = W[640 + u], w11 = W[704 + u];
  const float w12 = W[768 + u],   w13 = W[832 + u], w14 = W[896 + u];

  float* __restrict__ O = out + (size_t)z * ODIM;

  // real-CG coefficients (Frobenius-normalized tables * sqrt(2*lo+1)), closed form
  constexpr float IS2   = 0.70710678118654752f;  // 1/sqrt(2)
  constexpr float IS3   = 0.57735026918962576f;  // 1/sqrt(3)
  constexpr float IS5   = 0.44721359549995794f;  // 1/sqrt(5)
  constexpr float IS6   = 0.40824829046386302f;  // 1/sqrt(6)
  constexpr float IS10  = 0.31622776601683794f;  // 1/sqrt(10)
  constexpr float S3_10 = 0.54772255750516611f;  // sqrt(3/10)
  constexpr float S3    = 1.73205080756887729f;
  constexpr float A222  = 0.53452248382484879f;  // sqrt(2/7)
  constexpr float B222  = 0.26726124191242439f;  // sqrt(1/14)
  constexpr float C222  = 0.46291004988627577f;  // sqrt(3/14)

  // p0 (0,0,0) @0
  O[u] = w0 * s * vs;
  // p1 (0,1,1) @64
  { float* o = O + 64 + u * 3; const float c = w1 * s;
    o[0] = c * q0; o[1] = c * q1; o[2] = c * q2; }
  // p2 (0,2,2) @256
  { float* o = O + 256 + u * 5; const float c = w2 * s;
    o[0] = c * e0; o[1] = c * e1; o[2] = c * e2; o[3] = c * e3; o[4] = c * e4; }
  // p3 (1,0,1) @576
  { float* o = O + 576 + u * 3; const float c = w3 * vs;
    o[0] = c * p0; o[1] = c * p1; o[2] = c * p2; }
  // p4 (1,1,0) @768
  O[768 + u] = w4 * IS3 * (p0 * q0 + p1 * q1 + p2 * q2);
  // p5 (1,1,1) @832 : cross product / sqrt(2)
  { float* o = O + 832 + u * 3; const float c = w5 * IS2;
    o[0] = c * (p1 * q2 - p2 * q1);
    o[1] = c * (p2 * q0 - p0 * q2);
    o[2] = c * (p0 * q1 - p1 * q0); }
  // p6 (1,1,2) @1024 : symmetric traceless
  { float* o = O + 1024 + u * 5;
    o[0] = w6 * IS2 * (p0 * q2 + p2 * q0);
    o[1] = w6 * IS2 * (p0 * q1 + p1 * q0);
    o[2] = w6 * IS6 * (2.f * p1 * q1 - p2 * q2 - p0 * q0);
    o[3] = w6 * IS2 * (p1 * q2 + p2 * q1);
    o[4] = w6 * IS2 * (p2 * q2 - p0 * q0); }
  // p7 (1,2,1) @1344
  { float* o = O + 1344 + u * 3;
    o[0] = w7 * (S3_10 * (p2 * e0 + p1 * e1 - p0 * e4) - IS10 * p0 * e2);
    o[1] = w7 * (S3_10 * (p0 * e1 + p2 * e3) + 2.f * IS10 * p1 * e2);
    o[2] = w7 * (S3_10 * (p0 * e0 + p1 * e3 + p2 * e4) - IS10 * p2 * e2); }
  // p8 (1,2,2) @1536 : generator type (odd parity)
  { float* o = O + 1536 + u * 5;
    o[0] = w8 * IS6 * (p0 * e1 + 2.f * p1 * e4 - p2 * e3);
    o[1] = w8 * (IS6 * (p1 * e3 - p0 * e0 - p2 * e4) - IS2 * p2 * e2);
    o[2] = w8 * IS2 * (p2 * e1 - p0 * e3);
    o[3] = w8 * (IS2 * p0 * e2 + IS6 * (p2 * e0 - p0 * e4 - p1 * e1));
    o[4] = w8 * IS6 * (p0 * e3 - 2.f * p1 * e0 + p2 * e1); }
  // p9 (2,0,2) @1856
  { float* o = O + 1856 + u * 5; const float c = w9 * vs;
    o[0] = c * d0; o[1] = c * d1; o[2] = c * d2; o[3] = c * d3; o[4] = c * d4; }
  // p10 (2,1,1) @2176
  { float* o = O + 2176 + u * 3;
    o[0] = w10 * (S3_10 * (d0 * q2 + d1 * q1 - d4 * q0) - IS10 * d2 * q0);
    o[1] = w10 * (S3_10 * (d1 * q0 + d3 * q2) + 2.f * IS10 * d2 * q1);
    o[2] = w10 * (S3_10 * (d0 * q0 + d3 * q1 + d4 * q2) - IS10 * d2 * q2); }
  // p11 (2,1,2) @2368
  { float* o = O + 2368 + u * 5;
    o[0] = w11 * IS6 * (d1 * q0 + 2.f * d4 * q1 - d3 * q2);
    o[1] = w11 * (IS6 * (d3 * q1 - d0 * q0 - d4 * q2) - IS2 * d2 * q2);
    o[2] = w11 * IS2 * (d1 * q2 - d3 * q0);
    o[3] = w11 * (IS2 * d2 * q0 + IS6 * (d0 * q2 - d4 * q0 - d1 * q1));
    o[4] = w11 * IS6 * (d3 * q0 - 2.f * d0 * q1 + d1 * q2); }
  // p12 (2,2,0) @2688
  O[2688 + u] = w12 * IS5 * (d0 * e0 + d1 * e1 + d2 * e2 + d3 * e3 + d4 * e4);
  // p13 (2,2,1) @2752 : l=2 generators
  { float* o = O + 2752 + u * 3;
    o[0] = w13 * IS10 * (d1 * e0 - d0 * e1 + S3 * (d2 * e3 - d3 * e2) - d4 * e3 + d3 * e4);
    o[1] = w13 * IS10 * (2.f * (d4 * e0 - d0 * e4) + d3 * e1 - d1 * e3);
    o[2] = w13 * IS10 * (d0 * e3 - d3 * e0 + S3 * (d1 * e2 - d2 * e1) - d4 * e1 + d1 * e4); }
  // p14 (2,2,2) @2944
  { float* o = O + 2944 + u * 5;
    o[0] = w14 * (C222 * (d1 * e3 + d3 * e1) - A222 * (d2 * e0 + d0 * e2));
    o[1] = w14 * (B222 * (d2 * e1 + d1 * e2) + C222 * (d0 * e3 + d3 * e0 - d4 * e1 - d1 * e4));
    o[2] = w14 * (A222 * (d2 * e2 - d4 * e4 - d0 * e0) + B222 * (d1 * e1 + d3 * e3));
    o[3] = w14 * (B222 * (d2 * e3 + d3 * e2) + C222 * (d4 * e3 + d3 * e4 + d0 * e1 + d1 * e0));
    o[4] = w14 * (C222 * (d3 * e3 - d1 * e1) - A222 * (d2 * e4 + d4 * e2)); }
}

extern "C" void kernel_launch(void* const* d_in, const int* in_sizes, int n_in,
                              void* d_out, int out_size, void* d_ws, size_t ws_size,
                              hipStream_t stream) {
  const float* x1 = (const float*)d_in[0];
  const float* x2 = (const float*)d_in[1];
  const float* w  = (const float*)d_in[2];
  float* out = (float*)d_out;
  const int n = in_sizes[0] / DIM1;
  const int grid = (n + ZPB - 1) / ZPB;
  tp_kernel<<<grid, NTHR, 0, stream>>>(x1, x2, w, out, n);
}